// NegSNRLoss_84335977824805
// MI455X (gfx1250) — compile-verified
//
#include <hip/hip_runtime.h>

typedef __attribute__((ext_vector_type(2))) float v2f;
typedef __attribute__((ext_vector_type(8))) float v8f;

#define WMMA_F32(A, B, C) \
  __builtin_amdgcn_wmma_f32_16x16x4_f32(false, (A), false, (B), (short)0, (C), false, false)

// ---------------- problem constants (fixed by the reference) ----------------
constexpr int N_SAMP  = 16384;   // samples per row
constexpr int MIN_IDX = 364;     // argmin |f - 40/60|  on linspace(0,15,8193)
constexpr int MAX_IDX = 2276;    // argmin |f - 250/60|
constexpr int M_LO    = 5;       // k = s + 64*m, m in [5, 36] covers [364,2276)
// ---------------- LDS layout (floats) ----------------
// strides picked so every WMMA operand read is a conflict-free ds_load_b64
constexpr int X_STRIDE  = 68;                 // x as [q][p], q<256, p<64
constexpr int X_SIZE    = 256 * X_STRIDE;     // 17408 (later reused for B2: 64*260=16640)
constexpr int B1_STRIDE = 68;                 // DFT-64 table [c][p], c<128 (cos|sin)
constexpr int B1_SIZE   = 128 * B1_STRIDE;    // 8704
constexpr int Y_STRIDE  = 260;                // Y' as [s][q], s<64, q<256
constexpr int Y_SIZE    = 64 * Y_STRIDE;      // 16640
constexpr int B2_STRIDE = 260;                // DFT-256 table [j][q], j<64 (cos|sin)
constexpr int P1_SIZE   = 2048;               // power, indexed k-320
constexpr int RED_SIZE  = 256;
constexpr int TW_SIZE   = 256;                // 128 complex entries (c,s) interleaved
constexpr int LDS_FLOATS =
    X_SIZE + B1_SIZE + 2 * Y_SIZE + P1_SIZE + RED_SIZE + 2 * TW_SIZE;  // 62208

__global__ __launch_bounds__(256)
void snr_row_kernel(const float* __restrict__ x,
                    const float* __restrict__ targets,
                    float* __restrict__ snr_out) {
  extern __shared__ float lds[];
  float* xl  = lds;                 // phase 0/1: x[q][p]; phase 2: B2 table
  float* b1  = lds + X_SIZE;        // stage-1 DFT matrix [c][p]
  float* yr  = b1 + B1_SIZE;        // twiddled stage-1 real, [s][q]
  float* yi  = yr + Y_SIZE;         // twiddled stage-1 imag, [s][q]
  float* p1  = yi + Y_SIZE;         // power spectrum, index k-320
  float* red = p1 + P1_SIZE;
  float* twl = red + RED_SIZE;      // W_16384^j,   j<128, (cos, sin) pairs
  float* twh = twl + TW_SIZE;       // W_16384^128h = W_128^h, h<128

  const int tid  = threadIdx.x;
  const int wave = tid >> 5;
  const int lane = tid & 31;
  const int half = lane >> 4;
  const int ln   = lane & 15;
  const int row  = blockIdx.x;
  const float* __restrict__ xg = x + (size_t)row * N_SAMP;

  // -------- phase 0: stage row into LDS as [q][p] (n = 256p + q); build tables
  for (int i = tid; i < N_SAMP / 4; i += 256) {
    const float4 v = reinterpret_cast<const float4*>(xg)[i];
    const int n = i * 4;
    const int p = n >> 8;
    const int q = n & 255;
    xl[(q + 0) * X_STRIDE + p] = v.x;
    xl[(q + 1) * X_STRIDE + p] = v.y;
    xl[(q + 2) * X_STRIDE + p] = v.z;
    xl[(q + 3) * X_STRIDE + p] = v.w;
  }
  for (int i = tid; i < 64 * 64; i += 256) {
    const int p = i >> 6, s = i & 63;
    const float t = (float)((p * s) & 63) * (2.0f / 64.0f);
    b1[s * B1_STRIDE + p]        = cospif(t);   // C[p,s]
    b1[(64 + s) * B1_STRIDE + p] = sinpif(t);   // S[p,s]
  }
  if (tid < 128) {
    const float tl = (float)tid * (2.0f / 16384.0f);
    const float th = (float)tid * (2.0f / 128.0f);
    twl[2 * tid]     = cospif(tl);
    twl[2 * tid + 1] = sinpif(tl);
    twh[2 * tid]     = cospif(th);
    twh[2 * tid + 1] = sinpif(th);
  }
  __syncthreads();

  // -------- phase 1: stage-1 GEMM  Y[q,s] = sum_p x[256p+q] * W64^{sp}
  // Each wave: 2 q-tiles; per q-tile keep 8 accumulators (4 s-blocks x cos/sin)
  // so each K-step is 1 A-load + 8 B-loads feeding 8 independent WMMAs.
  for (int g = 0; g < 2; ++g) {
    const int qt = wave * 2 + g;
    v8f acc[8];
#pragma unroll
    for (int i = 0; i < 8; ++i) acc[i] = (v8f){};
#pragma unroll 2
    for (int kk = 0; kk < 16; ++kk) {
      const int p0 = kk * 4 + 2 * half;
      const v2f a = *(const v2f*)&xl[(qt * 16 + ln) * X_STRIDE + p0];
#pragma unroll
      for (int sb = 0; sb < 4; ++sb) {
        const v2f bc = *(const v2f*)&b1[(sb * 16 + ln) * B1_STRIDE + p0];
        const v2f bs = *(const v2f*)&b1[(64 + sb * 16 + ln) * B1_STRIDE + p0];
        acc[2 * sb]     = WMMA_F32(a, bc, acc[2 * sb]);
        acc[2 * sb + 1] = WMMA_F32(a, bs, acc[2 * sb + 1]);
      }
    }
    // twiddle by W_16384^{sq} (factored table lookup) and store transposed [s][q]
#pragma unroll
    for (int sb = 0; sb < 4; ++sb) {
#pragma unroll
      for (int r = 0; r < 8; ++r) {
        const int q  = qt * 16 + r + 8 * half;
        const int s  = sb * 16 + ln;
        const int a  = q * s;                       // < 16384
        const v2f wl = *(const v2f*)&twl[2 * (a & 127)];
        const v2f wh = *(const v2f*)&twh[2 * (a >> 7)];
        const float c  = wh.x * wl.x - wh.y * wl.y; // W = c - i*s
        const float sn = wh.x * wl.y + wh.y * wl.x;
        const float Yr = acc[2 * sb][r];
        const float Yi = -acc[2 * sb + 1][r];
        yr[s * Y_STRIDE + q] = Yr * c + Yi * sn;
        yi[s * Y_STRIDE + q] = Yi * c - Yr * sn;
      }
    }
  }
  __syncthreads();

  // -------- phase 1.5: build stage-2 DFT-256 table into (now free) x region
  float* b2 = xl;  // [j][q], j<32: cos(2*pi*m*q/256), j>=32: sin;  m = M_LO + (j&31)
  for (int i = tid; i < 32 * 256; i += 256) {
    const int j = i >> 8, q = i & 255;
    const float t = (float)(((M_LO + j) * q) & 255) * (2.0f / 256.0f);
    b2[j * B2_STRIDE + q]        = cospif(t);
    b2[(32 + j) * B2_STRIDE + q] = sinpif(t);
  }
  __syncthreads();

  // -------- phase 2: stage-2 complex GEMM  X[s+64m] = sum_q Y'[q,s] * W256^{mq}
  // 8 units (4 s-tiles x 2 m-tiles), one per wave; K = 256 as 64 WMMA steps
  {
    const int st = wave >> 1;
    const int mt = wave & 1;
    v8f xrA = {};
    v8f xiA = {};
#pragma unroll 2
    for (int kk = 0; kk < 64; ++kk) {
      const int q0 = kk * 4 + 2 * half;
      const v2f ar = *(const v2f*)&yr[(st * 16 + ln) * Y_STRIDE + q0];
      const v2f ai = *(const v2f*)&yi[(st * 16 + ln) * Y_STRIDE + q0];
      const v2f bc = *(const v2f*)&b2[(mt * 16 + ln) * B2_STRIDE + q0];
      const v2f bs = *(const v2f*)&b2[(32 + mt * 16 + ln) * B2_STRIDE + q0];
      const v2f nbs = -bs;                 // f32 WMMA has no A/B NEG modifier
      xrA = WMMA_F32(ar, bc, xrA);
      xiA = WMMA_F32(ai, bc, xiA);
      xrA = WMMA_F32(ai, bs, xrA);
      xiA = WMMA_F32(ar, nbs, xiA);
    }
#pragma unroll
    for (int r = 0; r < 8; ++r) {
      const int s = st * 16 + r + 8 * half;
      const int j = mt * 16 + ln;          // m = M_LO + j ; k = s + 64m ; idx = k - 320
      const float vr = xrA[r], vi = xiA[r];
      p1[j * 64 + s] = vr * vr + vi * vi;
    }
  }
  __syncthreads();

  // -------- phase 3: masked band sum, pulse bin, SNR (fixed-order => deterministic)
  const float t_hz = targets[row] * (1.0f / 60.0f);
  int ref = (int)floorf(t_hz * (8192.0f / 15.0f) + 0.5f);
  if (ref < MIN_IDX + 1) ref = MIN_IDX + 1;
  if (ref > MAX_IDX - 3) ref = MAX_IDX - 3;

  float part = 0.0f;
  for (int idx = tid; idx < 2048; idx += 256) {
    const int k = idx + 320;
    if (k >= MIN_IDX && k < MAX_IDX && (k < ref - 1 || k > ref + 1)) part += p1[idx];
  }
  red[tid] = part;
  __syncthreads();
  if (tid == 0) {
    float other = 0.0f;
    for (int i = 0; i < 256; ++i) other += red[i];
    const float pulse = p1[ref - 320];
    const float denom = (float)(MAX_IDX - MIN_IDX - 3);  // 1909
    snr_out[row] = 10.0f * log10f(pulse * denom / other);
  }
}

__global__ __launch_bounds__(256)
void snr_reduce_kernel(const float* __restrict__ snr, float* __restrict__ out, int rows) {
  __shared__ float red[256];
  float part = 0.0f;
  for (int i = threadIdx.x; i < rows; i += 256) part += snr[i];
  red[threadIdx.x] = part;
  __syncthreads();
  if (threadIdx.x == 0) {
    float s = 0.0f;
    for (int i = 0; i < 256; ++i) s += red[i];
    out[0] = -s / (float)rows;
  }
}

extern "C" void kernel_launch(void* const* d_in, const int* in_sizes, int n_in,
                              void* d_out, int out_size, void* d_ws, size_t ws_size,
                              hipStream_t stream) {
  const float* x       = (const float*)d_in[0];
  const float* targets = (const float*)d_in[1];
  float* snr = (float*)d_ws;                 // rows floats of scratch
  const int rows = in_sizes[0] / N_SAMP;
  const size_t lds_bytes = (size_t)LDS_FLOATS * sizeof(float);  // ~243 KB < 320 KB/WGP
  (void)hipFuncSetAttribute((const void*)snr_row_kernel,
                            hipFuncAttributeMaxDynamicSharedMemorySize,
                            (int)lds_bytes);
  snr_row_kernel<<<rows, 256, lds_bytes, stream>>>(x, targets, snr);
  snr_reduce_kernel<<<1, 256, 0, stream>>>(snr, (float*)d_out, rows);
}